// Simplex_10668698763828
// MI455X (gfx1250) — compile-verified
//
#include <hip/hip_runtime.h>

#define ROWLEN  8192
#define THREADS 256
#define NWAVES  (THREADS / 32)
#define VPT     (ROWLEN / (THREADS * 4))   // float4 chunks per thread = 8
#define NUM_ITERS 6

// Reduce a (sum, count) pair across the whole 256-thread block.
// Wave32 shuffle tree, then 8 wave partials combined through LDS.
__device__ __forceinline__ float2 block_reduce2(float2 val, float* redS, float* redC,
                                                int tid) {
    #pragma unroll
    for (int off = 16; off >= 1; off >>= 1) {
        val.x += __shfl_xor(val.x, off, 32);
        val.y += __shfl_xor(val.y, off, 32);
    }
    __syncthreads();                 // protect red[] against previous round's readers
    if ((tid & 31) == 0) {
        redS[tid >> 5] = val.x;
        redC[tid >> 5] = val.y;
    }
    __syncthreads();
    float2 acc = make_float2(0.f, 0.f);
    #pragma unroll
    for (int w = 0; w < NWAVES; ++w) {
        acc.x += redS[w];
        acc.y += redC[w];
    }
    return acc;
}

__global__ __launch_bounds__(THREADS) void simplex_rows_kernel(
        const float* __restrict__ x, float* __restrict__ out) {
    __shared__ float lds_row[ROWLEN];
    __shared__ float redS[NWAVES];
    __shared__ float redC[NWAVES];

    const int tid = threadIdx.x;
    const int row = blockIdx.x;
    const uint64_t rowbase = (uint64_t)(uintptr_t)(x + (size_t)row * ROWLEN);
    const uint32_t lds_base = (uint32_t)(uintptr_t)(&lds_row[0]);

    // ---- CDNA5 async DMA: stage the whole 32 KB row into LDS ----
    // 256 lanes x 8 x b128 = 32768 bytes, GVS mode (SGPR64 base + VGPR32 offset).
    #pragma unroll
    for (int c = 0; c < VPT; ++c) {
        const uint32_t eoff    = (uint32_t)(c * THREADS + tid) * 16u;  // byte offset
        const uint32_t ldsaddr = lds_base + eoff;
        asm volatile("global_load_async_to_lds_b128 %0, %1, %2"
                     :
                     : "v"(ldsaddr), "v"(eoff), "s"(rowbase)
                     : "memory");
    }
    asm volatile("s_wait_asynccnt 0" ::: "memory");
    __syncthreads();   // all 8 waves' DMA complete -> full row visible in LDS

    // ---- Pull this thread's 32 elements into registers (8 x ds_load_b128) ----
    float4 v[VPT];
    #pragma unroll
    for (int c = 0; c < VPT; ++c) {
        v[c] = *reinterpret_cast<const float4*>(&lds_row[(c * THREADS + tid) * 4]);
    }

    // ---- Round 0: mask = ones -> plain sum, count known (8192) ----
    float s0 = 0.f;
    #pragma unroll
    for (int c = 0; c < VPT; ++c)
        s0 += (v[c].x + v[c].y) + (v[c].z + v[c].w);
    float2 r = block_reduce2(make_float2(s0, 0.f), redS, redC, tid);
    float t = (r.x - 1.0f) / (float)ROWLEN;

    // ---- Rounds 1..5: masked sum + active count, mask = (x - t >= 0) ----
    #pragma unroll
    for (int it = 1; it < NUM_ITERS; ++it) {
        float s = 0.f, cnt = 0.f;
        #pragma unroll
        for (int c = 0; c < VPT; ++c) {
            if (v[c].x - t >= 0.f) { s += v[c].x; cnt += 1.f; }
            if (v[c].y - t >= 0.f) { s += v[c].y; cnt += 1.f; }
            if (v[c].z - t >= 0.f) { s += v[c].z; cnt += 1.f; }
            if (v[c].w - t >= 0.f) { s += v[c].w; cnt += 1.f; }
        }
        r = block_reduce2(make_float2(s, cnt), redS, redC, tid);
        t = (r.x - 1.0f) / r.y;
    }

    // ---- Output: x - t, coalesced b128 stores straight from registers ----
    float4* outp = reinterpret_cast<float4*>(out + (size_t)row * ROWLEN);
    #pragma unroll
    for (int c = 0; c < VPT; ++c) {
        float4 o = v[c];
        o.x -= t; o.y -= t; o.z -= t; o.w -= t;
        outp[c * THREADS + tid] = o;
    }
}

extern "C" void kernel_launch(void* const* d_in, const int* in_sizes, int n_in,
                              void* d_out, int out_size, void* d_ws, size_t ws_size,
                              hipStream_t stream) {
    const float* x = (const float*)d_in[0];
    float* out = (float*)d_out;
    (void)in_sizes; (void)n_in; (void)out_size; (void)d_ws; (void)ws_size;
    simplex_rows_kernel<<<ROWLEN, THREADS, 0, stream>>>(x, out);
}